// RuleNetwork_28235115004096
// MI455X (gfx1250) — compile-verified
//
#include <hip/hip_runtime.h>
#include <hip/hip_bf16.h>

#define B_  32768
#define V_  3
#define H_  512
#define R_  4
#define D_  64
#define DK_ 32
#define K_  (V_*H_)   /* 1536 */

typedef __attribute__((ext_vector_type(2))) float v2f;
typedef __attribute__((ext_vector_type(8))) float v8f;

__device__ __forceinline__ v8f wmma4(v2f a, v2f b, v8f c) {
  // D = A(16x4 f32) * B(4x16 f32) + C(16x16 f32)
  return __builtin_amdgcn_wmma_f32_16x16x4_f32(false, a, false, b, (short)0, c,
                                               false, false);
}

__device__ __forceinline__ float gumbel_noise(float u) {
  const float EPS = 1e-6f;
  return -logf(-logf(u * (1.0f - 2.0f * EPS) + EPS) + EPS);
}

// GEMM over one 512-wide K segment (one variable v = SEG).
// q accumulators accumulate across all segments; s accumulators only this one.
template<int SEG>
__device__ __forceinline__ void gemm_seg(
    const float* __restrict__ hr0, const float* __restrict__ hr1,
    const float* __restrict__ Wq, const float* __restrict__ Wk,
    int kh2, int lrow,
    v8f& q00, v8f& q01, v8f& q10, v8f& q11,
    v8f& s00, v8f& s01, v8f& s10, v8f& s11)
{
  const int kend = SEG * 512 + 512;
  for (int k = SEG * 512; k < kend; k += 4) {
    if ((k & 63) == 0 && (k + 256) < K_) {
      __builtin_prefetch(hr0 + k + 256, 0, 1);
      __builtin_prefetch(hr1 + k + 256, 0, 1);
    }
    // A 16x4 f32 layout: lane<16 holds A[M=lane,K=0..1]; lane>=16 holds K=2..3
    v2f a0 = *(const v2f*)(hr0 + k + kh2);
    v2f a1 = *(const v2f*)(hr1 + k + kh2);
    // B 4x16 f32 layout (mirrors C): vgpr0: K=(0|2), vgpr1: K=(1|3), N=lane&15
    const int kb = (k + kh2) * DK_;
    v2f bq0, bq1, bk0, bk1;
    bq0.x = Wq[kb + lrow];       bq0.y = Wq[kb + DK_ + lrow];
    bq1.x = Wq[kb + 16 + lrow];  bq1.y = Wq[kb + DK_ + 16 + lrow];
    bk0.x = Wk[kb + lrow];       bk0.y = Wk[kb + DK_ + lrow];
    bk1.x = Wk[kb + 16 + lrow];  bk1.y = Wk[kb + DK_ + 16 + lrow];
    q00 = wmma4(a0, bq0, q00);
    q01 = wmma4(a0, bq1, q01);
    q10 = wmma4(a1, bq0, q10);
    q11 = wmma4(a1, bq1, q11);
    s00 = wmma4(a0, bk0, s00);
    s01 = wmma4(a0, bk1, s01);
    s10 = wmma4(a1, bk0, s10);
    s11 = wmma4(a1, bk1, s11);
  }
}

__global__ __launch_bounds__(32)
void rulenet_fused(const float* __restrict__ hidden,
                   const float* __restrict__ rule_emb,
                   const float* __restrict__ Wq_r,
                   const float* __restrict__ bq_r,
                   const float* __restrict__ Wk_r,
                   const float* __restrict__ bk_r,
                   const float* __restrict__ Wq_v,
                   const float* __restrict__ Wk_v,
                   const float* __restrict__ Wp,
                   const float* __restrict__ bp,
                   const float* __restrict__ gu_rule,
                   const float* __restrict__ gu_var,
                   float* __restrict__ out)
{
  __shared__ float s_P[R_][H_];          // remb @ Wp              (8 KB)
  __shared__ float s_bp[H_];             //                        (2 KB)
  __shared__ float s_bqr[DK_];
  __shared__ float s_kr[R_][DK_];        // remb @ Wk_r + bk_r
  __shared__ float s_qv[R_][DK_];        // remb @ Wq_v
  __shared__ float s_qr[32][DK_ + 1];    // GEMM out: q_r rows (padded vs bank conflicts)
  __shared__ float s_kv[32][V_][DK_ + 1];// GEMM out: k_v rows
  __shared__ float s_mv[32];             // rule mask value at argmax
  __shared__ int   s_rr[32];             // argmax rule index
  __shared__ float s_vm[32][V_];         // var mask values

  const int lane = threadIdx.x;
  const int kh   = lane >> 4;            // half-wave id
  const int kh2  = kh * 2;
  const int lrow = lane & 15;
  const int rowbase = blockIdx.x * 32;
  const float inv_s = 0.17677669529663687f; // 1/sqrt(32)

  // ---------- Phase 0: batch-independent small matrices ----------
  for (int i = lane; i < R_ * H_; i += 32) {
    const int r = i >> 9, h = i & (H_ - 1);
    float acc = 0.f;
    #pragma unroll 8
    for (int d = 0; d < D_; ++d) acc += rule_emb[r * D_ + d] * Wp[d * H_ + h];
    s_P[r][h] = acc;
  }
  for (int i = lane; i < H_; i += 32) s_bp[i] = bp[i];
  if (lane < DK_) s_bqr[lane] = bq_r[lane];
  for (int i = lane; i < R_ * DK_; i += 32) {
    const int r = i >> 5, dk = i & (DK_ - 1);
    float a1 = bk_r[dk], a2 = 0.f;
    #pragma unroll 8
    for (int d = 0; d < D_; ++d) {
      const float re = rule_emb[r * D_ + d];
      a1 += re * Wk_r[d * DK_ + dk];
      a2 += re * Wq_v[d * DK_ + dk];
    }
    s_kr[r][dk] = a1;
    s_qv[r][dk] = a2;
  }
  __syncthreads();

  // ---------- Phase 1: WMMA GEMM  [32 rows x 1536] x ([1536x32] q | per-seg k_v) ----
  const float* hr0 = hidden + (size_t)(rowbase + lrow) * K_;
  const float* hr1 = hidden + (size_t)(rowbase + 16 + lrow) * K_;

  v8f q00 = {}, q01 = {}, q10 = {}, q11 = {};
  v8f k0_00 = {}, k0_01 = {}, k0_10 = {}, k0_11 = {};
  v8f k1_00 = {}, k1_01 = {}, k1_10 = {}, k1_11 = {};
  v8f k2_00 = {}, k2_01 = {}, k2_10 = {}, k2_11 = {};

  gemm_seg<0>(hr0, hr1, Wq_r, Wk_v, kh2, lrow, q00, q01, q10, q11,
              k0_00, k0_01, k0_10, k0_11);
  gemm_seg<1>(hr0, hr1, Wq_r, Wk_v, kh2, lrow, q00, q01, q10, q11,
              k1_00, k1_01, k1_10, k1_11);
  gemm_seg<2>(hr0, hr1, Wq_r, Wk_v, kh2, lrow, q00, q01, q10, q11,
              k2_00, k2_01, k2_10, k2_11);

  // ---------- Phase 1.5: dump C tiles to LDS ----------
  // C layout: vgpr j, lane<16 -> M=j, N=lane ; lane>=16 -> M=j+8, N=lane-16
  #pragma unroll
  for (int j = 0; j < 8; ++j) {
    const int mlo = kh * 8 + j;
    const int mhi = 16 + kh * 8 + j;
    const int nlo = lrow;
    const int nhi = 16 + lrow;
    s_qr[mlo][nlo] = q00[j];  s_qr[mlo][nhi] = q01[j];
    s_qr[mhi][nlo] = q10[j];  s_qr[mhi][nhi] = q11[j];
    s_kv[mlo][0][nlo] = k0_00[j];  s_kv[mlo][0][nhi] = k0_01[j];
    s_kv[mhi][0][nlo] = k0_10[j];  s_kv[mhi][0][nhi] = k0_11[j];
    s_kv[mlo][1][nlo] = k1_00[j];  s_kv[mlo][1][nhi] = k1_01[j];
    s_kv[mhi][1][nlo] = k1_10[j];  s_kv[mhi][1][nhi] = k1_11[j];
    s_kv[mlo][2][nlo] = k2_00[j];  s_kv[mlo][2][nhi] = k2_01[j];
    s_kv[mhi][2][nlo] = k2_10[j];  s_kv[mhi][2][nhi] = k2_11[j];
  }
  __syncthreads();

  // ---------- Phase 2: per-row gumbel selection (lane == local row) ----------
  float* out_rule = out + (size_t)B_ * V_ * H_;
  float* out_var  = out_rule + (size_t)B_ * R_;
  {
    const int b = rowbase + lane;

    // rule scores + gumbel
    float z[R_];
    #pragma unroll
    for (int r = 0; r < R_; ++r) {
      float acc = 0.f;
      #pragma unroll 8
      for (int dk = 0; dk < DK_; ++dk)
        acc += (s_qr[lane][dk] + s_bqr[dk]) * s_kr[r][dk];
      z[r] = acc * inv_s + gumbel_noise(gu_rule[b * R_ + r]);
    }
    int am = 0; float zm = z[0];
    #pragma unroll
    for (int r = 1; r < R_; ++r) if (z[r] > zm) { zm = z[r]; am = r; }
    float y[R_], es = 0.f;
    #pragma unroll
    for (int r = 0; r < R_; ++r) { y[r] = expf(z[r] - zm); es += y[r]; }
    const float inv = 1.f / es;
    float mv = 0.f;
    #pragma unroll
    for (int r = 0; r < R_; ++r) {
      y[r] *= inv;
      const float hard = (r == am) ? 1.f : 0.f;
      const float m = (hard + y[r]) - y[r];     // straight-through forward value
      out_rule[b * R_ + r] = m;
      if (r == am) mv = m;
    }
    s_rr[lane] = am;
    s_mv[lane] = mv;

    // variable scores + gumbel
    float zv[V_];
    #pragma unroll
    for (int v = 0; v < V_; ++v) {
      float acc = 0.f;
      #pragma unroll 8
      for (int dk = 0; dk < DK_; ++dk)
        acc += (mv * s_qv[am][dk]) * s_kv[lane][v][dk];
      zv[v] = acc * inv_s + gumbel_noise(gu_var[b * V_ + v]);
    }
    int av = 0; float zvm = zv[0];
    #pragma unroll
    for (int v = 1; v < V_; ++v) if (zv[v] > zvm) { zvm = zv[v]; av = v; }
    float yv[V_], esv = 0.f;
    #pragma unroll
    for (int v = 0; v < V_; ++v) { yv[v] = expf(zv[v] - zvm); esv += yv[v]; }
    const float invv = 1.f / esv;
    #pragma unroll
    for (int v = 0; v < V_; ++v) {
      yv[v] *= invv;
      const float hard = (v == av) ? 1.f : 0.f;
      const float m = (hard + yv[v]) - yv[v];
      out_var[b * V_ + v] = m;
      s_vm[lane][v] = m;
    }
  }
  __syncthreads();

  // ---------- Phase 3: streamed output  out = hidden + vm * (mv*P[r*] + bp) ----------
  for (int row = 0; row < 32; ++row) {
    const int   rr = s_rr[row];
    const float mv = s_mv[row];
    const float4* hin  = (const float4*)(hidden + (size_t)(rowbase + row) * K_);
    float4*       hout = (float4*)(out + (size_t)(rowbase + row) * K_);
    #pragma unroll 4
    for (int i = lane; i < K_ / 4; i += 32) {
      const int idx = i * 4;
      const int v = idx >> 9;
      const int h = idx & (H_ - 1);
      const float  vm = s_vm[row][v];
      float4 x        = hin[i];
      const float4 p  = *(const float4*)&s_P[rr][h];
      const float4 bb = *(const float4*)&s_bp[h];
      x.x += vm * (mv * p.x + bb.x);
      x.y += vm * (mv * p.y + bb.y);
      x.z += vm * (mv * p.z + bb.z);
      x.w += vm * (mv * p.w + bb.w);
      hout[i] = x;
    }
  }
}

extern "C" void kernel_launch(void* const* d_in, const int* in_sizes, int n_in,
                              void* d_out, int out_size, void* d_ws, size_t ws_size,
                              hipStream_t stream) {
  (void)in_sizes; (void)n_in; (void)d_ws; (void)ws_size; (void)out_size;
  const float* hidden   = (const float*)d_in[0];
  const float* rule_emb = (const float*)d_in[1];
  const float* Wq_r     = (const float*)d_in[2];
  const float* bq_r     = (const float*)d_in[3];
  const float* Wk_r     = (const float*)d_in[4];
  const float* bk_r     = (const float*)d_in[5];
  const float* Wq_v     = (const float*)d_in[6];
  const float* Wk_v     = (const float*)d_in[7];
  const float* Wp       = (const float*)d_in[8];
  const float* bp       = (const float*)d_in[9];
  const float* gu_rule  = (const float*)d_in[10];
  const float* gu_var   = (const float*)d_in[11];
  float* out = (float*)d_out;

  dim3 grid(B_ / 32), block(32);
  rulenet_fused<<<grid, block, 0, stream>>>(hidden, rule_emb, Wq_r, bq_r, Wk_r,
                                            bk_r, Wq_v, Wk_v, Wp, bp, gu_rule,
                                            gu_var, out);
}